// VanillaRNN_30623116821278
// MI455X (gfx1250) — compile-verified
//
#include <hip/hip_runtime.h>
#include <hip/hip_bf16.h>

#define B_SZ 1024
#define T_SZ 1024
#define H_SZ 256
#define C_SZ 10

typedef __attribute__((ext_vector_type(2))) float v2f;
typedef __attribute__((ext_vector_type(8))) float v8f;

// Hardware tanh (gfx1250 TRANS op); fallback via exp2/rcp which also hit TRANS.
__device__ __forceinline__ float fast_tanh(float v) {
#if __has_builtin(__builtin_amdgcn_tanhf)
    return __builtin_amdgcn_tanhf(v);
#else
    // tanh(v) = 1 - 2/(exp(2v)+1);  2*log2(e) = 2.885390081777927
    float e = __builtin_amdgcn_exp2f(v * 2.885390081777927f);
    return 1.0f - 2.0f * __builtin_amdgcn_rcpf(e + 1.0f);
#endif
}

// One block per batch row b. Thread h owns hidden unit h's serial chain.
// x[b, :] is staged through LDS in 256-element chunks; every lane reads the
// same LDS word per step (bank broadcast, no conflicts).
__global__ __launch_bounds__(256) void rnn_scan_kernel(
    const float* __restrict__ x,      // [B, T]
    const float* __restrict__ W_hx,   // [H, 1]
    const float* __restrict__ W_hh,   // [H, H] (only diagonal used)
    const float* __restrict__ b_h,    // [H]
    float* __restrict__ hbuf)         // [B, H] out
{
    const int b = blockIdx.x;
    const int h = threadIdx.x;

    const float win = W_hx[h];
    const float wd  = W_hh[(size_t)h * (H_SZ + 1)];  // diagonal element
    const float bh  = b_h[h];

    const float* xrow = x + (size_t)b * T_SZ;
    __shared__ float xs[256];

    float hs = 0.0f;
    for (int t0 = 0; t0 < T_SZ; t0 += 256) {
        __syncthreads();
        xs[h] = xrow[t0 + h];
        if (t0 + 256 < T_SZ)
            __builtin_prefetch(xrow + t0 + 256 + h, 0, 0);  // global_prefetch_b8
        __syncthreads();
#pragma unroll 16
        for (int i = 0; i < 256; ++i) {
            // h = tanh(x_t*w_in + h*w_diag + b_h)
            hs = fast_tanh(__builtin_fmaf(xs[i], win, __builtin_fmaf(hs, wd, bh)));
        }
    }
    hbuf[(size_t)b * H_SZ + h] = hs;
}

// Final projection: out[b,c] = sum_h hbuf[b,h] * W_hp[c,h] + b_o[c].
// One wave (32 lanes) per 16-row tile of the output, full-precision fp32 WMMA:
//   D(16x16) += A(16x4) x B(4x16), K-loop over H=256, N padded 10 -> 16.
// A layout (32b, 16x4): lanes 0-15 hold M=lane, {K=0 in v0, K=1 in v1};
//                       lanes 16-31 hold M=lane-16, {K=2, K=3}.
// B layout mirrors A with N in place of M. C/D: VGPR r = row r (+8 for hi lanes).
__global__ __launch_bounds__(32) void rnn_proj_kernel(
    const float* __restrict__ hbuf,   // [B, H]
    const float* __restrict__ W_hp,   // [C, H]
    const float* __restrict__ b_o,    // [C]
    float* __restrict__ out)          // [B, C]
{
    const int lane = threadIdx.x;
    const int b0   = blockIdx.x * 16;
    const int mn   = lane & 15;   // A row index == B col index for this lane
    const int kg   = lane >> 4;   // K sub-group (0: K+0/1, 1: K+2/3)

    const float* hrow = hbuf + (size_t)(b0 + mn) * H_SZ;
    const int   ncl   = (mn < C_SZ) ? mn : 0;       // stay in-bounds
    const float msk   = (mn < C_SZ) ? 1.0f : 0.0f;  // zero-pad N=10..15
    const float* wrow = W_hp + (size_t)ncl * H_SZ;

    v8f acc = {};
#pragma unroll 4
    for (int k0 = 0; k0 < H_SZ; k0 += 4) {
        const int k = k0 + 2 * kg;
        v2f a;  a.x  = hrow[k];        a.y  = hrow[k + 1];
        v2f bm; bm.x = wrow[k] * msk;  bm.y = wrow[k + 1] * msk;
        // 8 args: (neg_a, A, neg_b, B, c_mod, C, reuse_a, reuse_b)
        acc = __builtin_amdgcn_wmma_f32_16x16x4_f32(
            false, a, false, bm, (short)0, acc, false, false);
    }

    if (mn < C_SZ) {
        const float bo = b_o[mn];
#pragma unroll
        for (int r = 0; r < 8; ++r) {
            const int row = b0 + r + 8 * kg;
            out[(size_t)row * C_SZ + mn] = acc[r] + bo;
        }
    }
}

extern "C" void kernel_launch(void* const* d_in, const int* in_sizes, int n_in,
                              void* d_out, int out_size, void* d_ws, size_t ws_size,
                              hipStream_t stream) {
    const float* x    = (const float*)d_in[0];   // [B, T]
    const float* W_hx = (const float*)d_in[1];   // [H, 1]
    const float* W_hh = (const float*)d_in[2];   // [H, H]
    const float* b_h  = (const float*)d_in[3];   // [H]
    const float* W_hp = (const float*)d_in[4];   // [C, H]
    const float* b_o  = (const float*)d_in[5];   // [C]
    float* out  = (float*)d_out;                 // [B, C]
    float* hbuf = (float*)d_ws;                  // B*H floats = 1 MiB scratch

    rnn_scan_kernel<<<B_SZ, 256, 0, stream>>>(x, W_hx, W_hh, b_h, hbuf);
    rnn_proj_kernel<<<B_SZ / 16, 32, 0, stream>>>(hbuf, W_hp, b_o, out);
}